// SpatialFusion_55439437857537
// MI455X (gfx1250) — compile-verified
//
#include <hip/hip_runtime.h>
#include <math.h>

// ---------------------------------------------------------------------------
// Segment-max fusion: out[b, :] = max over rows x[off[b] .. off[b+1]) of x.
// Pure HBM-streaming kernel: 553.6 MB in + 138.4 MB out, no reuse ->
// non-temporal b128 loads/stores, 30 us roofline at 23.3 TB/s.
// ---------------------------------------------------------------------------

typedef __attribute__((ext_vector_type(4))) float v4f;

static __device__ __forceinline__ v4f v4max(v4f a, v4f b) {
    v4f r;
    r.x = __builtin_fmaxf(a.x, b.x);
    r.y = __builtin_fmaxf(a.y, b.y);
    r.z = __builtin_fmaxf(a.z, b.z);
    r.w = __builtin_fmaxf(a.w, b.w);
    return r;
}

// Prologue: exclusive prefix sum of record_len -> offsets[0..B] in workspace.
// B is tiny (8); one lane does it.
__global__ void seg_offsets_kernel(const int* __restrict__ record_len, int B,
                                   int* __restrict__ offsets) {
    if (blockIdx.x == 0 && threadIdx.x == 0) {
        int acc = 0;
        offsets[0] = 0;
        for (int b = 0; b < B; ++b) {
            acc += record_len[b];
            offsets[b + 1] = acc;
        }
    }
}

// Main kernel (vectorized): grid = (ceil(per_map4/256), B).
// per_map4 = (C*H*W)/4 float4 elements per feature map.
__global__ __launch_bounds__(256) void seg_max_v4_kernel(
    const float* __restrict__ x, const int* __restrict__ offsets,
    float* __restrict__ out, unsigned per_map4) {
    unsigned inner = blockIdx.x * blockDim.x + threadIdx.x;
    if (inner >= per_map4) return;
    unsigned b = blockIdx.y;

    int s = offsets[b];
    int e = offsets[b + 1];

    const v4f* __restrict__ xv = (const v4f*)x;
    v4f* __restrict__ ov = (v4f*)out;

    v4f acc;
    acc.x = -INFINITY; acc.y = -INFINITY; acc.z = -INFINITY; acc.w = -INFINITY;

    int cnt = e - s;
    size_t base = (size_t)s * (size_t)per_map4 + (size_t)inner;

    if (cnt == 4) {
        // Specialized path: issue all 4 streaming b128 loads back-to-back
        // (max memory-level parallelism), then reduce.
        v4f a0 = __builtin_nontemporal_load(&xv[base]);
        v4f a1 = __builtin_nontemporal_load(&xv[base + (size_t)per_map4]);
        v4f a2 = __builtin_nontemporal_load(&xv[base + 2ull * per_map4]);
        v4f a3 = __builtin_nontemporal_load(&xv[base + 3ull * per_map4]);
        acc = v4max(v4max(a0, a1), v4max(a2, a3));
    } else {
        for (int n = 0; n < cnt; ++n) {
            v4f a = __builtin_nontemporal_load(&xv[base + (size_t)n * per_map4]);
            acc = v4max(acc, a);
        }
    }

    __builtin_nontemporal_store(acc, &ov[(size_t)b * per_map4 + inner]);
}

// Scalar fallback for per-map sizes not divisible by 4 (not hit with the
// reference shapes, but keeps the kernel correct for arbitrary C*H*W).
__global__ __launch_bounds__(256) void seg_max_s_kernel(
    const float* __restrict__ x, const int* __restrict__ offsets,
    float* __restrict__ out, unsigned per_map) {
    unsigned inner = blockIdx.x * blockDim.x + threadIdx.x;
    if (inner >= per_map) return;
    unsigned b = blockIdx.y;

    int s = offsets[b];
    int e = offsets[b + 1];

    float acc = -INFINITY;
    size_t base = (size_t)s * (size_t)per_map + (size_t)inner;
    for (int n = 0; n < e - s; ++n) {
        float a = __builtin_nontemporal_load(&x[base + (size_t)n * per_map]);
        acc = __builtin_fmaxf(acc, a);
    }
    __builtin_nontemporal_store(acc, &out[(size_t)b * per_map + inner]);
}

extern "C" void kernel_launch(void* const* d_in, const int* in_sizes, int n_in,
                              void* d_out, int out_size, void* d_ws, size_t ws_size,
                              hipStream_t stream) {
    const float* x          = (const float*)d_in[0];
    const int*   record_len = (const int*)d_in[1];
    // d_in[2] = loc_flg, d_in[3] = loca_inference_reverse_flg: the reference
    // implements only the loc_flg==0 segment-max branch, so they are unused.

    float* out = (float*)d_out;

    const int B = in_sizes[1];              // number of segments
    const int per_map = out_size / B;       // C*H*W elements per feature map

    int* offsets = (int*)d_ws;              // needs (B+1) ints of scratch

    seg_offsets_kernel<<<1, 32, 0, stream>>>(record_len, B, offsets);

    if ((per_map & 3) == 0) {
        const unsigned per_map4 = (unsigned)(per_map >> 2);
        dim3 grid((per_map4 + 255u) / 256u, (unsigned)B);
        seg_max_v4_kernel<<<grid, 256, 0, stream>>>(x, offsets, out, per_map4);
    } else {
        dim3 grid(((unsigned)per_map + 255u) / 256u, (unsigned)B);
        seg_max_s_kernel<<<grid, 256, 0, stream>>>(x, offsets, out, (unsigned)per_map);
    }
}